// HighOrderActivation_841813590310
// MI455X (gfx1250) — compile-verified
//
#include <hip/hip_runtime.h>
#include <hip/hip_bf16.h>

typedef float v2f __attribute__((ext_vector_type(2)));
typedef float v8f __attribute__((ext_vector_type(8)));

#define B_TOT 2048
#define D_TOT 256
#define K_X   5
#define P_N   32   // 2^K rows of params per d
#define O_N   32   // output features
#define WAVES 8
#define ROWS_PER_WAVE 32
#define ROWS_PER_BLOCK (ROWS_PER_WAVE * WAVES)  // 256
#define WPAD 33    // padded LDS row stride (floats) to avoid bank conflicts

__global__ __launch_bounds__(256)
void HighOrderActivation_wmma_kernel(const float* __restrict__ X,
                                     const float* __restrict__ P,
                                     float* __restrict__ out) {
    __shared__ float p_lds[P_N][WPAD];                       // params[d] : 32x32
    __shared__ float w_lds[WAVES][ROWS_PER_WAVE][WPAD];      // dense weights per wave

    const int tid  = threadIdx.x;
    const int wave = tid >> 5;
    const int lane = tid & 31;
    const int d    = blockIdx.y;
    const int b0   = blockIdx.x * ROWS_PER_BLOCK;

    // ---- Stage 1: params[d] (32x32 f32 = 4KB) -> LDS ----
    for (int i = tid; i < P_N * O_N; i += 256) {
        const int p = i >> 5, o = i & 31;
        p_lds[p][o] = P[((size_t)d * P_N + p) * O_N + o];
    }

    // ---- Stage 2: per-lane row -> sort 5, build dense w[32] in LDS ----
    {
        const int b = b0 + wave * ROWS_PER_WAVE + lane;
        const float* xr = X + ((size_t)b * D_TOT + d) * K_X;
        float v[K_X];
        int   id[K_X];
#pragma unroll
        for (int k = 0; k < K_X; ++k) { v[k] = xr[k]; id[k] = k; }

        // 9-comparator sorting network for n=5 (ascending), tracking argsort
#define CSWAP(i, j)                                                        \
        { if (v[i] > v[j]) { float tv = v[i]; v[i] = v[j]; v[j] = tv;      \
                             int   ti = id[i]; id[i] = id[j]; id[j] = ti; } }
        CSWAP(0,1) CSWAP(3,4) CSWAP(2,4) CSWAP(2,3) CSWAP(1,4)
        CSWAP(0,3) CSWAP(0,2) CSWAP(1,3) CSWAP(1,2)
#undef CSWAP

        // zero the dense weight row
#pragma unroll
        for (int p = 0; p < P_N; ++p) w_lds[wave][lane][p] = 0.0f;

        // reverse cumulative bit-sum -> pidx (strictly decreasing => distinct)
        int pidx[K_X];
        int acc = 0;
#pragma unroll
        for (int k = K_X - 1; k >= 0; --k) { acc += (1 << id[k]); pidx[k] = acc; }

        // first differences of sorted values -> scatter into w
        float coef[K_X];
        coef[0] = v[0];
#pragma unroll
        for (int k = 1; k < K_X; ++k) coef[k] = v[k] - v[k - 1];
#pragma unroll
        for (int k = 0; k < K_X; ++k) w_lds[wave][lane][pidx[k]] = coef[k];
    }

    __syncthreads();

    // ---- Stage 3: per-wave (32x32) @ (32x32) GEMM via V_WMMA_F32_16X16X4_F32 ----
    const int half = lane >> 4;   // 0: lanes 0-15 (K+0/1), 1: lanes 16-31 (K+2/3)
    const int l15  = lane & 15;

#pragma unroll
    for (int mt = 0; mt < 2; ++mt) {
#pragma unroll
        for (int nt = 0; nt < 2; ++nt) {
            v8f acc = {0.f, 0.f, 0.f, 0.f, 0.f, 0.f, 0.f, 0.f};
#pragma unroll
            for (int k0 = 0; k0 < P_N; k0 += 4) {
                const int ka = k0 + half * 2;
                v2f a, bb;
                // A 16x4 fragment: lane -> M (0-15), VGPR0/1 -> K pair
                a.x  = w_lds[wave][mt * 16 + l15][ka + 0];
                a.y  = w_lds[wave][mt * 16 + l15][ka + 1];
                // B 4x16 fragment: lane -> N (0-15), VGPR0/1 -> K pair
                bb.x = p_lds[ka + 0][nt * 16 + l15];
                bb.y = p_lds[ka + 1][nt * 16 + l15];
                // 8 args: (neg_a, A, neg_b, B, c_mod, C, reuse_a, reuse_b)
                acc = __builtin_amdgcn_wmma_f32_16x16x4_f32(
                    false, a, false, bb, (short)0, acc, false, false);
            }
            // ---- Stage 4: store D per ISA layout (VGPR r: M=r / M=r+8) ----
            const int col = nt * 16 + l15;
#pragma unroll
            for (int r = 0; r < 8; ++r) {
                const int m = mt * 16 + half * 8 + r;
                const int b = b0 + wave * ROWS_PER_WAVE + m;
                out[((size_t)b * D_TOT + d) * O_N + col] = acc[r];
            }
        }
    }
}

extern "C" void kernel_launch(void* const* d_in, const int* in_sizes, int n_in,
                              void* d_out, int out_size, void* d_ws, size_t ws_size,
                              hipStream_t stream) {
    (void)in_sizes; (void)n_in; (void)out_size; (void)d_ws; (void)ws_size;
    const float* X = (const float*)d_in[0];      // (2048, 256, 5) f32
    const float* P = (const float*)d_in[1];      // (256, 32, 32) f32
    float* out = (float*)d_out;                  // (2048, 256, 32) f32

    dim3 grid(B_TOT / ROWS_PER_BLOCK, D_TOT);    // (8, 256)
    HighOrderActivation_wmma_kernel<<<grid, 256, 0, stream>>>(X, P, out);
}